// GCN_42434276884780
// MI455X (gfx1250) — compile-verified
//
#include <hip/hip_runtime.h>
#include <hip/hip_bf16.h>
#include <math.h>

#define N_NODES  100000
#define N_EDGES  1600000
#define N_GRAPHS 512

typedef __attribute__((ext_vector_type(2))) float v2f;
typedef __attribute__((ext_vector_type(8))) float v8f;

// ---- order-preserving float <-> uint encoding for atomic max pooling ----
__device__ __forceinline__ unsigned enc_f(float f) {
    unsigned u = __float_as_uint(f);
    return (u & 0x80000000u) ? ~u : (u | 0x80000000u);
}
__device__ __forceinline__ float dec_f(unsigned e) {
    unsigned u = (e & 0x80000000u) ? (e & 0x7fffffffu) : ~e;
    return __uint_as_float(u);
}

// ---- degree accumulation: one thread per edge ----
__global__ __launch_bounds__(256) void deg_kernel(const int* __restrict__ src,
                                                  const int* __restrict__ dst,
                                                  float* __restrict__ degOut,
                                                  float* __restrict__ degIn, int n) {
    int i = blockIdx.x * blockDim.x + threadIdx.x;
    if (i < n) {
        atomicAdd(&degOut[src[i]], 1.0f);
        atomicAdd(&degIn[dst[i]], 1.0f);
    }
}

// ---- deg -> rsqrt(max(deg,1)) in place ----
__global__ __launch_bounds__(256) void rsqrt_kernel(float* __restrict__ a, int n) {
    int i = blockIdx.x * blockDim.x + threadIdx.x;
    if (i < n) a[i] = rsqrtf(fmaxf(a[i], 1.0f));
}

// ---- xs = x * norm_s (row-wise, 64 features) ----
__global__ __launch_bounds__(256) void rowscale64_kernel(const float* __restrict__ x,
                                                         const float* __restrict__ s,
                                                         float* __restrict__ y, int n) {
    int i = blockIdx.x * blockDim.x + threadIdx.x;
    if (i < n) y[i] = x[i] * s[i >> 6];
}

// ---- edge aggregation: out[dst] += hs[src], D features, float4 per thread ----
template <int D>
__global__ __launch_bounds__(256) void edge_agg_kernel(const float* __restrict__ hs,
                                                       const int* __restrict__ src,
                                                       const int* __restrict__ dst,
                                                       float* __restrict__ out) {
    constexpr int C = D / 4;  // float4 chunks per edge
    long long tid = (long long)blockIdx.x * blockDim.x + threadIdx.x;
    if (tid >= (long long)N_EDGES * C) return;
    int e  = (int)(tid / C);
    int ch = (int)(tid - (long long)e * C);
    const float4 v = *reinterpret_cast<const float4*>(hs + (size_t)src[e] * D + ch * 4);
    float* o = out + (size_t)dst[e] * D + ch * 4;
    atomicAdd(o + 0, v.x);
    atomicAdd(o + 1, v.y);
    atomicAdd(o + 2, v.z);
    atomicAdd(o + 3, v.w);
}

// ---- pool buffer init to enc(-inf) ----
__global__ __launch_bounds__(256) void pool_init_kernel(unsigned* __restrict__ pool, int n) {
    int i = blockIdx.x * blockDim.x + threadIdx.x;
    if (i < n) pool[i] = 0x007FFFFFu;  // enc(-inf)
}

// ---- WMMA f32 GEMM: D = (A * normd[row]) @ W + bias, tiled 16x16 per wave.
//      EPI=0: relu, *posts[row], store to out.  EPI=1: atomicMax pool by graph id.
template <int K, int EPI>
__global__ __launch_bounds__(256) void gemm_wmma_kernel(const float* __restrict__ A,
                                                        const float* __restrict__ normd,
                                                        const float* __restrict__ W,
                                                        const float* __restrict__ bias,
                                                        const float* __restrict__ posts,
                                                        const int* __restrict__ gids,
                                                        float* __restrict__ out,
                                                        unsigned* __restrict__ pool) {
    constexpr int N = 128;
    const int lane = threadIdx.x & 31;
    const int wave = threadIdx.x >> 5;         // 8 waves: nt = wave
    const int row0 = blockIdx.x * 16;          // mt = blockIdx.x
    const int col0 = wave * 16;
    const int l15 = lane & 15;
    const int kh  = (lane >> 4) * 2;           // 0 for lanes 0-15, 2 for lanes 16-31

    const int arow = row0 + l15;
    const float nd = normd[arow];
    const float* Ap = A + (size_t)arow * K + kh;
    const float* Wp = W + (size_t)kh * N + col0 + l15;

    v8f c = {};
#pragma unroll
    for (int k = 0; k < K; k += 4) {
        // A fragment 16x4: lane<16 holds A[m][k],A[m][k+1]; lane>=16 holds A[m][k+2],A[m][k+3]
        v2f a;
        a.x = Ap[0] * nd;
        a.y = Ap[1] * nd;
        // B fragment 4x16: lane<16 holds W[k][n],W[k+1][n]; lane>=16 holds W[k+2][n],W[k+3][n]
        v2f b;
        b.x = Wp[0];
        b.y = Wp[N];
        c = __builtin_amdgcn_wmma_f32_16x16x4_f32(false, a, false, b, (short)0, c, false, false);
        Ap += 4;
        Wp += 4 * N;
    }

    // C/D layout: VGPR r, lanes 0-15 -> M=r, lanes 16-31 -> M=r+8; N = lane&15
    const int nidx = col0 + l15;
    const int rbase = row0 + (lane >> 4) * 8;
    const float bv = bias[nidx];
    if (EPI == 0) {
#pragma unroll
        for (int r = 0; r < 8; r++) {
            int row = rbase + r;
            float v = c[r] + bv;
            v = fmaxf(v, 0.0f) * posts[row];
            out[(size_t)row * N + nidx] = v;
        }
    } else {
#pragma unroll
        for (int r = 0; r < 8; r++) {
            int row = rbase + r;
            float v = c[r] + bv;
            atomicMax(&pool[(size_t)gids[row] * N + nidx], enc_f(v));
        }
    }
}

// ---- readout: out[g] = where(isfinite(max), max, 0) @ Wl + bl ----
__global__ __launch_bounds__(256) void final_linear_kernel(const unsigned* __restrict__ pool,
                                                           const float* __restrict__ Wl,
                                                           const float* __restrict__ bl,
                                                           float* __restrict__ out) {
    __shared__ float hg[128];
    const int g = blockIdx.x;
    const int t = threadIdx.x;
    if (t < 128) {
        float v = dec_f(pool[g * 128 + t]);
        hg[t] = isfinite(v) ? v : 0.0f;
    }
    __syncthreads();
    if (t < 225) {
        float acc = bl[t];
#pragma unroll 4
        for (int k = 0; k < 128; k++) acc += hg[k] * Wl[k * 225 + t];
        out[g * 225 + t] = acc;
    }
}

extern "C" void kernel_launch(void* const* d_in, const int* in_sizes, int n_in,
                              void* d_out, int out_size, void* d_ws, size_t ws_size,
                              hipStream_t stream) {
    const float* x   = (const float*)d_in[0];
    const int*   src = (const int*)d_in[1];
    const int*   dst = (const int*)d_in[2];
    const int*   gid = (const int*)d_in[3];
    const float* W1  = (const float*)d_in[4];
    const float* b1  = (const float*)d_in[5];
    const float* W2  = (const float*)d_in[6];
    const float* b2  = (const float*)d_in[7];
    const float* Wl  = (const float*)d_in[8];
    const float* bl  = (const float*)d_in[9];
    float* out = (float*)d_out;

    // workspace carve-up (256B aligned)
    char* ws = (char*)d_ws;
    size_t off = 0;
    auto carve = [&](size_t bytes) -> void* {
        void* p = ws + off;
        off += (bytes + 255) & ~(size_t)255;
        return p;
    };
    float*    norm_s  = (float*)carve((size_t)N_NODES * 4);
    float*    norm_d  = (float*)carve((size_t)N_NODES * 4);
    float*    regionA = (float*)carve((size_t)N_NODES * 128 * 4);  // xs+agg1, later agg2
    float*    h1s     = (float*)carve((size_t)N_NODES * 128 * 4);
    unsigned* pool    = (unsigned*)carve((size_t)N_GRAPHS * 128 * 4);

    float* xs   = regionA;
    float* agg1 = regionA + (size_t)N_NODES * 64;
    float* agg2 = regionA;  // reused after GEMM1 (xs/agg1 dead by then)

    // degrees -> norms
    hipMemsetAsync(norm_s, 0, (size_t)N_NODES * 4, stream);
    hipMemsetAsync(norm_d, 0, (size_t)N_NODES * 4, stream);
    deg_kernel<<<(N_EDGES + 255) / 256, 256, 0, stream>>>(src, dst, norm_s, norm_d, N_EDGES);
    rsqrt_kernel<<<(N_NODES + 255) / 256, 256, 0, stream>>>(norm_s, N_NODES);
    rsqrt_kernel<<<(N_NODES + 255) / 256, 256, 0, stream>>>(norm_d, N_NODES);

    // layer 1: xs = x*norm_s ; agg1 = scatter-add ; h1s = relu(agg1*norm_d @ W1 + b1) * norm_s
    {
        int n = N_NODES * 64;
        rowscale64_kernel<<<(n + 255) / 256, 256, 0, stream>>>(x, norm_s, xs, n);
    }
    hipMemsetAsync(agg1, 0, (size_t)N_NODES * 64 * 4, stream);
    {
        long long total = (long long)N_EDGES * 16;
        edge_agg_kernel<64><<<(int)((total + 255) / 256), 256, 0, stream>>>(xs, src, dst, agg1);
    }
    gemm_wmma_kernel<64, 0><<<N_NODES / 16, 256, 0, stream>>>(agg1, norm_d, W1, b1, norm_s,
                                                              nullptr, h1s, nullptr);

    // layer 2: agg2 = scatter-add of h1s ; pool = segment_max(agg2*norm_d @ W2 + b2)
    hipMemsetAsync(agg2, 0, (size_t)N_NODES * 128 * 4, stream);
    {
        long long total = (long long)N_EDGES * 32;
        edge_agg_kernel<128><<<(int)((total + 255) / 256), 256, 0, stream>>>(h1s, src, dst, agg2);
    }
    pool_init_kernel<<<(N_GRAPHS * 128 + 255) / 256, 256, 0, stream>>>(pool, N_GRAPHS * 128);
    gemm_wmma_kernel<128, 1><<<N_NODES / 16, 256, 0, stream>>>(agg2, norm_d, W2, b2, nullptr,
                                                               gid, nullptr, pool);

    // readout
    final_linear_kernel<<<N_GRAPHS, 256, 0, stream>>>(pool, Wl, bl, out);
}